// trainableBiquad4_47330539602257
// MI455X (gfx1250) — compile-verified
//
#include <hip/hip_runtime.h>
#include <cstdint>
#include <cstddef>

// ---------------------------------------------------------------------------
// trainableBiquad4: 4th-order IIR (two cascaded biquads) over [B,S,D], then
// y = a*y + c.  Memory-bound (~200MB traffic -> ~8.6us floor @ 23.3TB/s).
// Parallelized over channels AND sequence chunks via the linear-state
// decomposition  s_end = A^L * s_start + d  (A = per-channel 4x4).
// Phase 3 uses gfx1250 async global->LDS loads + s_wait_asynccnt pipelining.
// ---------------------------------------------------------------------------

constexpr int Bn = 8, Sn = 4096, Dn = 768;
constexpr int CHUNKS = 16;            // sequence chunks
constexpr int LCH    = Sn / CHUNKS;   // 256 steps per chunk
constexpr int LOG2L  = 8;             // log2(LCH)
constexpr int TPB    = 256;           // threads per block (8 waves, wave32)
constexpr int DT     = Dn / TPB;      // 3 channel tiles
constexpr int SUB    = 16;            // time-rows staged per async stage
constexpr int NSTG   = LCH / SUB;     // 16 stages per chunk

#if defined(__has_builtin)
#if __has_builtin(__builtin_amdgcn_global_load_async_to_lds_b32)
#define HAVE_ASYNC_LDS 1
#endif
#endif

// builtin signature (from clang diagnostic): (as1 int*, as3 int*, imm, imm)
#define AS_GLOBAL(p) ((__attribute__((address_space(1))) int*)(p))
#define AS_LOCAL(p)  ((__attribute__((address_space(3))) int*)(p))

// ---------------------------------------------------------------------------
// Phase 1: per (b, chunk, d): run the cascade over the chunk with ZERO initial
// y-state, record final state d_i = (y1p, y1pp, y2p, y2pp).
// ---------------------------------------------------------------------------
__global__ __launch_bounds__(TPB)
void biquad4_phase1(const float* __restrict__ x,
                    const float* __restrict__ b1, const float* __restrict__ fa1,
                    const float* __restrict__ b2, const float* __restrict__ fa2,
                    float* __restrict__ dstates /* 4 planes of CHUNKS*B*D */) {
  const int tid = threadIdx.x;
  const int blk = blockIdx.x;
  const int dt  = blk % DT;
  const int ch  = (blk / DT) % CHUNKS;
  const int b   = blk / (DT * CHUNKS);
  const int d   = dt * TPB + tid;

  const float b10 = b1[0*Dn+d], b11 = b1[1*Dn+d], b12 = b1[2*Dn+d];
  const float a11 = fa1[0*Dn+d], a12 = fa1[1*Dn+d];
  const float b20 = b2[0*Dn+d], b21 = b2[1*Dn+d], b22 = b2[2*Dn+d];
  const float a21 = fa2[0*Dn+d], a22 = fa2[1*Dn+d];

  const int t0 = ch * LCH;
  const float* xb = x + (size_t)b * Sn * Dn + d;   // column base; row t at xb[t*Dn]
  float x1 = (t0 >= 1) ? xb[(size_t)(t0-1)*Dn] : 0.f;
  float x2 = (t0 >= 2) ? xb[(size_t)(t0-2)*Dn] : 0.f;
  float y1p = 0.f, y1pp = 0.f, y2p = 0.f, y2pp = 0.f;

#pragma unroll 4
  for (int t = t0; t < t0 + LCH; ++t) {
    const float xt = xb[(size_t)t * Dn];
    const float u1 = fmaf(b10, xt, fmaf(b11, x1, b12 * x2));
    const float y1 = fmaf(-a11, y1p, fmaf(-a12, y1pp, u1));
    const float y2 = fmaf(b20, y1, fmaf(b21, y1p,
                     fmaf(b22, y1pp, fmaf(-a21, y2p, -a22 * y2pp))));
    x2 = x1; x1 = xt; y1pp = y1p; y1p = y1; y2pp = y2p; y2p = y2;
  }

  const size_t P = (size_t)CHUNKS * Bn * Dn;
  const size_t o = ((size_t)ch * Bn + b) * Dn + d;
  dstates[0*P+o] = y1p; dstates[1*P+o] = y1pp;
  dstates[2*P+o] = y2p; dstates[3*P+o] = y2pp;
}

// ---------------------------------------------------------------------------
// Phase 2: per (b,d) channel: M = A^LCH by repeated squaring; sequentially
// propagate chunk start states:  s_{i+1} = M*s_i + d_i.  Stores s_start per
// chunk.  (Poles are tiny -> M ~ 0, numerically benign.)
// ---------------------------------------------------------------------------
__global__ __launch_bounds__(TPB)
void biquad4_phase2(const float* __restrict__ fa1,
                    const float* __restrict__ b2, const float* __restrict__ fa2,
                    const float* __restrict__ dstates,
                    float* __restrict__ sstart) {
  const int gid = blockIdx.x * TPB + threadIdx.x;    // gid = b*Dn + d
  const int d = gid % Dn;
  const float a11 = fa1[d], a12 = fa1[Dn+d];
  const float b20 = b2[d], b21 = b2[Dn+d], b22 = b2[2*Dn+d];
  const float a21 = fa2[d], a22 = fa2[Dn+d];

  // State s = (y1[t-1], y1[t-2], y2[t-1], y2[t-2]);  s' = A*s + input
  float M[4][4] = {
    { -a11,                  -a12,                  0.f,   0.f },
    {  1.f,                   0.f,                  0.f,   0.f },
    { fmaf(-b20, a11, b21),  fmaf(-b20, a12, b22), -a21,  -a22 },
    {  0.f,                   0.f,                  1.f,   0.f }};

  for (int it = 0; it < LOG2L; ++it) {              // M = M*M, 8x -> A^256
    float T[4][4];
#pragma unroll
    for (int i = 0; i < 4; ++i)
#pragma unroll
      for (int j = 0; j < 4; ++j) {
        float s = 0.f;
#pragma unroll
        for (int k = 0; k < 4; ++k) s = fmaf(M[i][k], M[k][j], s);
        T[i][j] = s;
      }
#pragma unroll
    for (int i = 0; i < 4; ++i)
#pragma unroll
      for (int j = 0; j < 4; ++j) M[i][j] = T[i][j];
  }

  const size_t P = (size_t)CHUNKS * Bn * Dn;
  float s0 = 0.f, s1 = 0.f, s2 = 0.f, s3 = 0.f;
  for (int ch = 0; ch < CHUNKS; ++ch) {
    const size_t o = (size_t)ch * Bn * Dn + gid;
    sstart[0*P+o] = s0; sstart[1*P+o] = s1; sstart[2*P+o] = s2; sstart[3*P+o] = s3;
    const float d0 = dstates[0*P+o], d1 = dstates[1*P+o];
    const float d2 = dstates[2*P+o], d3 = dstates[3*P+o];
    const float n0 = fmaf(M[0][0],s0, fmaf(M[0][1],s1, fmaf(M[0][2],s2, fmaf(M[0][3],s3, d0))));
    const float n1 = fmaf(M[1][0],s0, fmaf(M[1][1],s1, fmaf(M[1][2],s2, fmaf(M[1][3],s3, d1))));
    const float n2 = fmaf(M[2][0],s0, fmaf(M[2][1],s1, fmaf(M[2][2],s2, fmaf(M[2][3],s3, d2))));
    const float n3 = fmaf(M[3][0],s0, fmaf(M[3][1],s1, fmaf(M[3][2],s2, fmaf(M[3][3],s3, d3))));
    s0 = n0; s1 = n1; s2 = n2; s3 = n3;
  }
}

// ---------------------------------------------------------------------------
// Phase 3: per (b, chunk, d): re-run the cascade with the correct start state
// and write out = a*y2 + c.  x tile is staged into LDS with gfx1250 async
// global->LDS copies, double buffered, pipelined with s_wait_asynccnt.
// Each thread consumes only its own staged column -> no barriers needed.
// ---------------------------------------------------------------------------
__global__ __launch_bounds__(TPB)
void biquad4_phase3(const float* __restrict__ x,
                    const float* __restrict__ a, const float* __restrict__ c,
                    const float* __restrict__ b1, const float* __restrict__ fa1,
                    const float* __restrict__ b2, const float* __restrict__ fa2,
                    const float* __restrict__ sstart,
                    float* __restrict__ out) {
  __shared__ float sx[2][SUB * TPB];                // 2 x 16KB double buffer

  const int tid = threadIdx.x;
  const int blk = blockIdx.x;
  const int dt  = blk % DT;
  const int ch  = (blk / DT) % CHUNKS;
  const int b   = blk / (DT * CHUNKS);
  const int d   = dt * TPB + tid;

  const float b10 = b1[0*Dn+d], b11 = b1[1*Dn+d], b12 = b1[2*Dn+d];
  const float a11 = fa1[0*Dn+d], a12 = fa1[1*Dn+d];
  const float b20 = b2[0*Dn+d], b21 = b2[1*Dn+d], b22 = b2[2*Dn+d];
  const float a21 = fa2[0*Dn+d], a22 = fa2[1*Dn+d];
  const float av = a[0], cv = c[0];

  const size_t P = (size_t)CHUNKS * Bn * Dn;
  const size_t o = ((size_t)ch * Bn + b) * Dn + d;
  float y1p = sstart[0*P+o], y1pp = sstart[1*P+o];
  float y2p = sstart[2*P+o], y2pp = sstart[3*P+o];

  const int t0 = ch * LCH;
  const float* xb = x + (size_t)b * Sn * Dn + d;
  float x1 = (t0 >= 1) ? xb[(size_t)(t0-1)*Dn] : 0.f;
  float x2 = (t0 >= 2) ? xb[(size_t)(t0-2)*Dn] : 0.f;
  float* ob = out + (size_t)b * Sn * Dn + d;

  // thread's staging column in the tile
  const float* gcol = x + ((size_t)b * Sn + t0) * Dn + dt * TPB + tid;

#ifdef HAVE_ASYNC_LDS
  auto prefetch = [&](int s) {
    const float* g = gcol + (size_t)s * SUB * Dn;
    float* l = &sx[s & 1][tid];
#pragma unroll
    for (int r = 0; r < SUB; ++r)
      __builtin_amdgcn_global_load_async_to_lds_b32(
          AS_GLOBAL(g + (size_t)r * Dn), AS_LOCAL(l + r * TPB), 0, 0);
  };
  prefetch(0);
#endif

  for (int s = 0; s < NSTG; ++s) {
#ifdef HAVE_ASYNC_LDS
    if (s + 1 < NSTG) {
      prefetch(s + 1);                              // keep next stage in flight
      asm volatile("s_wait_asynccnt %0" :: "i"(SUB) : "memory");   // stage s done
    } else {
      asm volatile("s_wait_asynccnt 0" ::: "memory");
    }
#else
#pragma unroll
    for (int r = 0; r < SUB; ++r)
      sx[s & 1][r * TPB + tid] = gcol[((size_t)s * SUB + r) * Dn];
#endif

#pragma unroll
    for (int r = 0; r < SUB; ++r) {
      const float xt = sx[s & 1][r * TPB + tid];
      const float u1 = fmaf(b10, xt, fmaf(b11, x1, b12 * x2));
      const float y1 = fmaf(-a11, y1p, fmaf(-a12, y1pp, u1));
      const float y2 = fmaf(b20, y1, fmaf(b21, y1p,
                       fmaf(b22, y1pp, fmaf(-a21, y2p, -a22 * y2pp))));
      x2 = x1; x1 = xt; y1pp = y1p; y1p = y1; y2pp = y2p; y2p = y2;
      ob[(size_t)(t0 + s * SUB + r) * Dn] = fmaf(av, y2, cv);
    }
  }
}

// ---------------------------------------------------------------------------
extern "C" void kernel_launch(void* const* d_in, const int* in_sizes, int n_in,
                              void* d_out, int out_size, void* d_ws, size_t ws_size,
                              hipStream_t stream) {
  const float* x   = (const float*)d_in[0];
  const float* a   = (const float*)d_in[1];
  const float* c   = (const float*)d_in[2];
  const float* b1  = (const float*)d_in[3];
  const float* fa1 = (const float*)d_in[4];
  const float* b2  = (const float*)d_in[5];
  const float* fa2 = (const float*)d_in[6];
  float* out = (float*)d_out;

  float* ws = (float*)d_ws;
  const size_t P = (size_t)CHUNKS * Bn * Dn;
  float* dstates = ws;            // 4*P floats
  float* sstart  = ws + 4 * P;    // 4*P floats   (total ~3.1 MB)

  dim3 blkDim(TPB);
  biquad4_phase1<<<Bn * CHUNKS * DT, blkDim, 0, stream>>>(x, b1, fa1, b2, fa2, dstates);
  biquad4_phase2<<<(Bn * Dn) / TPB, blkDim, 0, stream>>>(fa1, b2, fa2, dstates, sstart);
  biquad4_phase3<<<Bn * CHUNKS * DT, blkDim, 0, stream>>>(x, a, c, b1, fa1, b2, fa2,
                                                          sstart, out);
}